// FCOSPostProcessor_71081708749430
// MI455X (gfx1250) — compile-verified
//
#include <hip/hip_runtime.h>
#include <hip/hip_bf16.h>
#include <stdint.h>

// ---------------- problem constants (from reference) ----------------
#define N_IMG   8
#define NCLS    80
#define H_DIM   100
#define W_DIM   100
#define HW      (H_DIM * W_DIM)      // 10000
#define TOT     (NCLS * HW)          // 800000 per image
#define PRE_K   1000                 // PRE_NMS_TOP_N
#define POST_K  100                  // POST_NMS_TOP_N
#define NMS_TH  0.6f
#define CLS_OFF 4096.0f
#define PRE_TH  0.05f

// ---------------- implementation constants ----------------
#define NBINS       2048             // float-bits >> 19 for s in (0,1)
#define CAP         6144             // compaction capacity per image
#define CHUNK       1024             // hw positions per block tile
#define NCHUNK      ((HW + CHUNK - 1) / CHUNK)   // 10
#define CLS_GRP     4                // class groups per image
#define CLS_PER_BLK (NCLS / CLS_GRP) // 20
#define NSORT       8192             // bitonic sort size (64KB LDS)
#define FLT_MAXV    3.402823466e38f

__device__ __forceinline__ float sigm(float x) {
    return 1.0f / (1.0f + expf(-x));
}

// ---- CDNA5 async global->LDS copy (ASYNCcnt path), 16B per lane ----
// Generic pointers into LDS carry the wave-relative LDS byte offset in
// their low 32 bits (flat aperture mapping), which is exactly what the
// VDST operand of global_load_async_to_lds_* wants.
__device__ __forceinline__ void async_copy_b128(const float* gsrc, const float* ldst) {
    unsigned loff = (unsigned)(unsigned long long)(const void*)ldst;
    unsigned long long ga = (unsigned long long)(const void*)gsrc;
    asm volatile("global_load_async_to_lds_b128 %0, %1, off"
                 :: "v"(loff), "v"(ga) : "memory");
}
__device__ __forceinline__ void wait_async_le1() {
    asm volatile("s_wait_asynccnt 0x1" ::: "memory");
}
__device__ __forceinline__ void wait_async_0() {
    asm volatile("s_wait_asynccnt 0x0" ::: "memory");
}

// =====================================================================
// Kernel 1: sigmoid + threshold + score histogram (streaming, async LDS)
// grid = (NCHUNK, CLS_GRP, N_IMG), block = 256
// dynamic LDS: buf[2][CHUNK] + ctrs[CHUNK] + hist[NBINS]
// =====================================================================
__global__ void __launch_bounds__(256)
k_hist(const float* __restrict__ cls, const float* __restrict__ ctr,
       uint32_t* __restrict__ g_hist)
{
    extern __shared__ float smem[];
    float*    buf  = smem;                       // 2 * CHUNK floats
    float*    ctrs = smem + 2 * CHUNK;           // CHUNK floats
    uint32_t* hist = (uint32_t*)(smem + 3 * CHUNK); // NBINS uints

    const int n   = blockIdx.z;
    const int c0  = blockIdx.y * CLS_PER_BLK;
    const int hw0 = blockIdx.x * CHUNK;
    const int cnt = min(CHUNK, HW - hw0);        // 1024 or 784 (both %4==0)
    const int tid = threadIdx.x;
    const int j0  = tid * 4;
    const bool act = (j0 < cnt);

    for (int b = tid; b < NBINS; b += blockDim.x) hist[b] = 0;

    if (act) {
        const float* cbase = ctr + (size_t)n * HW + hw0;
        #pragma unroll
        for (int u = 0; u < 4; ++u) ctrs[j0 + u] = sigm(cbase[j0 + u]);
    }
    __syncthreads();

    const float* gbase = cls + (size_t)n * TOT + (size_t)c0 * HW + hw0;

    if (act) async_copy_b128(gbase + j0, buf + j0);            // prime c=0
    for (int c = 0; c < CLS_PER_BLK; ++c) {
        const int sel = c & 1;
        if (c + 1 < CLS_PER_BLK) {
            if (act) async_copy_b128(gbase + (size_t)(c + 1) * HW + j0,
                                     buf + (sel ^ 1) * CHUNK + j0);
            wait_async_le1();                                  // buf[sel] ready
        } else {
            wait_async_0();
        }
        if (act) {
            #pragma unroll
            for (int u = 0; u < 4; ++u) {
                const int j = j0 + u;
                const float sg = sigm(buf[sel * CHUNK + j]);
                if (sg > PRE_TH) {
                    const float s = sg * ctrs[j];
                    atomicAdd(&hist[__float_as_uint(s) >> 19], 1u);
                }
            }
        }
    }
    __syncthreads();
    for (int b = tid; b < NBINS; b += blockDim.x) {
        const uint32_t h = hist[b];
        if (h) atomicAdd(&g_hist[n * NBINS + b], h);
    }
}

// =====================================================================
// Kernel 2: per-image bin threshold from histogram suffix sum
// =====================================================================
__global__ void k_select(const uint32_t* __restrict__ g_hist,
                         uint32_t* __restrict__ tb)
{
    const int n = blockIdx.x;
    if (threadIdx.x != 0) return;
    unsigned acc = 0;
    int chosen = 0;
    for (int b = NBINS - 1; b >= 0; --b) {
        const unsigned c = g_hist[n * NBINS + b];
        acc += c;
        if (acc >= PRE_K) {
            chosen = (acc > CAP) ? (b + 1) : b;   // overflow guard (rare)
            break;
        }
    }
    tb[n] = (unsigned)chosen;
}

// =====================================================================
// Kernel 3: compaction of (score, flat_idx) above bin threshold
// grid = (NCHUNK, CLS_GRP, N_IMG), block = 256 (same async staging)
// =====================================================================
__global__ void __launch_bounds__(256)
k_compact(const float* __restrict__ cls, const float* __restrict__ ctr,
          const uint32_t* __restrict__ tb, uint32_t* __restrict__ cntg,
          float* __restrict__ comp_s, uint32_t* __restrict__ comp_i)
{
    extern __shared__ float smem[];
    float* buf  = smem;                 // 2 * CHUNK
    float* ctrs = smem + 2 * CHUNK;     // CHUNK

    const int n   = blockIdx.z;
    const int c0  = blockIdx.y * CLS_PER_BLK;
    const int hw0 = blockIdx.x * CHUNK;
    const int cnt = min(CHUNK, HW - hw0);
    const int tid = threadIdx.x;
    const int j0  = tid * 4;
    const bool act = (j0 < cnt);
    const unsigned tbin = tb[n];

    if (act) {
        const float* cbase = ctr + (size_t)n * HW + hw0;
        #pragma unroll
        for (int u = 0; u < 4; ++u) ctrs[j0 + u] = sigm(cbase[j0 + u]);
    }

    const float* gbase = cls + (size_t)n * TOT + (size_t)c0 * HW + hw0;

    if (act) async_copy_b128(gbase + j0, buf + j0);
    for (int c = 0; c < CLS_PER_BLK; ++c) {
        const int sel = c & 1;
        if (c + 1 < CLS_PER_BLK) {
            if (act) async_copy_b128(gbase + (size_t)(c + 1) * HW + j0,
                                     buf + (sel ^ 1) * CHUNK + j0);
            wait_async_le1();
        } else {
            wait_async_0();
        }
        if (act) {
            #pragma unroll
            for (int u = 0; u < 4; ++u) {
                const int j = j0 + u;
                const float sg = sigm(buf[sel * CHUNK + j]);
                if (sg > PRE_TH) {
                    const float s = sg * ctrs[j];
                    if ((__float_as_uint(s) >> 19) >= tbin) {
                        const unsigned pos = atomicAdd(&cntg[n], 1u);
                        if (pos < CAP) {
                            comp_s[n * CAP + pos] = s;
                            // flat index into (HW, C) layout: hw*C + c
                            comp_i[n * CAP + pos] =
                                (unsigned)((hw0 + j) * NCLS + (c0 + c));
                        }
                    }
                }
            }
        }
    }
}

// =====================================================================
// Kernel 4: exact top-1000 via in-LDS bitonic sort (desc score, asc idx)
// grid = N_IMG, block = 1024, dynamic LDS = NSORT*8 = 64KB
// =====================================================================
__global__ void __launch_bounds__(1024)
k_sort(const uint32_t* __restrict__ cntg, const float* __restrict__ comp_s,
       const uint32_t* __restrict__ comp_i, float* __restrict__ topv,
       int* __restrict__ topi)
{
    extern __shared__ unsigned long long keys[];
    const int n = blockIdx.x;
    const unsigned m = min(cntg[n], (unsigned)CAP);

    for (unsigned i = threadIdx.x; i < NSORT; i += blockDim.x) {
        unsigned long long k = 0ull;
        if (i < m) {
            const unsigned sb = __float_as_uint(comp_s[n * CAP + i]); // s>0
            const unsigned ix = comp_i[n * CAP + i];
            k = ((unsigned long long)sb << 32) |
                (unsigned long long)(0xFFFFFFFFu - ix);  // ties: smaller idx first
        }
        keys[i] = k;
    }
    __syncthreads();

    for (unsigned kk = 2; kk <= NSORT; kk <<= 1) {
        for (unsigned j = kk >> 1; j > 0; j >>= 1) {
            for (unsigned i = threadIdx.x; i < NSORT; i += blockDim.x) {
                const unsigned p = i ^ j;
                if (p > i) {
                    const unsigned long long a = keys[i], b = keys[p];
                    const bool descSeg = ((i & kk) == 0);
                    if (descSeg ? (a < b) : (a > b)) { keys[i] = b; keys[p] = a; }
                }
            }
            __syncthreads();
        }
    }

    for (unsigned i = threadIdx.x; i < PRE_K; i += blockDim.x) {
        const unsigned long long k = keys[i];
        topv[n * PRE_K + i] = __uint_as_float((unsigned)(k >> 32));
        topi[n * PRE_K + i] = (k == 0ull)
                                ? 0
                                : (int)(0xFFFFFFFFu - (unsigned)(k & 0xFFFFFFFFull));
    }
}

// =====================================================================
// Kernel 5: box decode + clip + class-offset greedy NMS + output
// grid = N_IMG, block = 1024
// =====================================================================
__global__ void __launch_bounds__(1024)
k_nms(const float* __restrict__ topv, const int* __restrict__ topi,
      const float* __restrict__ locations, const float* __restrict__ reg,
      const int* __restrict__ imsz, float* __restrict__ out)
{
    __shared__ float ox1[PRE_K], oy1[PRE_K], ox2[PRE_K], oy2[PRE_K]; // offset boxes
    __shared__ float bx1[PRE_K], by1[PRE_K], bx2[PRE_K], by2[PRE_K]; // clipped boxes
    __shared__ float vsc[PRE_K], lbl[PRE_K];
    __shared__ int   alive[PRE_K];
    __shared__ float rv[1024];
    __shared__ int   ri[1024];
    __shared__ int   keep[POST_K];
    __shared__ float cb[4];
    __shared__ int   cidx, cok;

    const int n = blockIdx.x;
    const int tid = threadIdx.x;
    const float hf = (float)imsz[2 * n + 0];
    const float wf = (float)imsz[2 * n + 1];

    if (tid < PRE_K) {
        float v = topv[n * PRE_K + tid];
        int idx = topi[n * PRE_K + tid];
        if (!(v > 0.0f)) { idx = 0; v = 0.0f; }      // invalid -> score 0
        const int loc = idx / NCLS;
        const int c1  = idx % NCLS + 1;               // label = class + 1
        const float px = locations[2 * loc + 0];
        const float py = locations[2 * loc + 1];
        const float* rg = reg + (size_t)n * 4 * HW;
        const float l = rg[0 * HW + loc], t = rg[1 * HW + loc];
        const float r = rg[2 * HW + loc], b = rg[3 * HW + loc];
        float x1 = px - l, y1 = py - t, x2 = px + r, y2 = py + b;
        x1 = fminf(fmaxf(x1, 0.0f), wf - 1.0f);
        y1 = fminf(fmaxf(y1, 0.0f), hf - 1.0f);
        x2 = fminf(fmaxf(x2, 0.0f), wf - 1.0f);
        y2 = fminf(fmaxf(y2, 0.0f), hf - 1.0f);
        const float off = (float)c1 * CLS_OFF;        // per-class NMS offset
        bx1[tid] = x1; by1[tid] = y1; bx2[tid] = x2; by2[tid] = y2;
        ox1[tid] = x1 + off; oy1[tid] = y1 + off;
        ox2[tid] = x2 + off; oy2[tid] = y2 + off;
        vsc[tid] = v;                                 // MIN_SIZE==0 -> always kept
        lbl[tid] = (float)c1;
        alive[tid] = 1;
    }
    __syncthreads();

    for (int it = 0; it < POST_K; ++it) {
        // argmax over alive entries, first occurrence wins ties
        if (tid < PRE_K && alive[tid]) { rv[tid] = vsc[tid]; ri[tid] = tid; }
        else { rv[tid] = -FLT_MAXV; ri[tid] = PRE_K; }
        __syncthreads();
        for (int s2 = 512; s2 > 0; s2 >>= 1) {
            if (tid < s2) {
                const float a = rv[tid], b = rv[tid + s2];
                const int ia = ri[tid], ib = ri[tid + s2];
                if (b > a || (b == a && ib < ia)) { rv[tid] = b; ri[tid] = ib; }
            }
            __syncthreads();
        }
        if (tid == 0) {
            const int idx = ri[0];
            const int ok = (rv[0] > 0.0f) ? 1 : 0;
            keep[it] = ok ? idx : -1;
            cidx = idx; cok = ok;
            if (idx < PRE_K) {
                cb[0] = ox1[idx]; cb[1] = oy1[idx];
                cb[2] = ox2[idx]; cb[3] = oy2[idx];
            }
        }
        __syncthreads();
        const int idx = cidx, ok = cok;
        if (tid < PRE_K) {
            if (!ok) {
                alive[tid] = 0;
            } else {
                const float xx1 = fmaxf(cb[0], ox1[tid]);
                const float yy1 = fmaxf(cb[1], oy1[tid]);
                const float xx2 = fminf(cb[2], ox2[tid]);
                const float yy2 = fminf(cb[3], oy2[tid]);
                const float inter = fmaxf(xx2 - xx1, 0.0f) * fmaxf(yy2 - yy1, 0.0f);
                const float a = fmaxf(cb[2] - cb[0], 0.0f) * fmaxf(cb[3] - cb[1], 0.0f);
                const float b = fmaxf(ox2[tid] - ox1[tid], 0.0f) *
                                fmaxf(oy2[tid] - oy1[tid], 0.0f);
                const float iou = inter / (a + b - inter + 1e-9f);
                if (iou > NMS_TH) alive[tid] = 0;
                if (tid == idx) alive[tid] = 0;
            }
        }
        __syncthreads();
    }

    if (tid < POST_K) {
        const int k = keep[tid];
        float* o = out + ((size_t)n * POST_K + tid) * 6;
        if (k >= 0) {
            o[0] = bx1[k]; o[1] = by1[k]; o[2] = bx2[k]; o[3] = by2[k];
            o[4] = vsc[k]; o[5] = lbl[k];
        } else {
            o[0] = 0.0f; o[1] = 0.0f; o[2] = 0.0f;
            o[3] = 0.0f; o[4] = 0.0f; o[5] = 0.0f;
        }
    }
}

// =====================================================================
extern "C" void kernel_launch(void* const* d_in, const int* in_sizes, int n_in,
                              void* d_out, int out_size, void* d_ws, size_t ws_size,
                              hipStream_t stream)
{
    (void)in_sizes; (void)n_in; (void)out_size; (void)ws_size;
    const float* locations = (const float*)d_in[0];   // [HW, 2]
    const float* box_cls   = (const float*)d_in[1];   // [N, C, H, W]
    const float* box_reg   = (const float*)d_in[2];   // [N, 4, H, W]
    const float* ctr       = (const float*)d_in[3];   // [N, 1, H, W]
    const int*   imsz      = (const int*)d_in[4];     // [N, 2]
    float* out = (float*)d_out;                       // [N, POST_K, 6]

    uint32_t* hist   = (uint32_t*)d_ws;               // N*NBINS
    uint32_t* cntg   = hist + N_IMG * NBINS;          // N
    uint32_t* tb     = cntg + N_IMG;                  // N
    float*    comp_s = (float*)(tb + N_IMG);          // N*CAP
    uint32_t* comp_i = (uint32_t*)(comp_s + N_IMG * CAP);
    float*    topv   = (float*)(comp_i + N_IMG * CAP);
    int*      topi   = (int*)(topv + N_IMG * PRE_K);

    hipMemsetAsync(d_ws, 0, (size_t)(N_IMG * NBINS + N_IMG) * sizeof(uint32_t),
                   stream);

    const dim3 gscan(NCHUNK, CLS_GRP, N_IMG);
    k_hist<<<gscan, 256, (3 * CHUNK + NBINS) * 4, stream>>>(box_cls, ctr, hist);
    k_select<<<N_IMG, 32, 0, stream>>>(hist, tb);
    k_compact<<<gscan, 256, (3 * CHUNK) * 4, stream>>>(box_cls, ctr, tb, cntg,
                                                       comp_s, comp_i);
    k_sort<<<N_IMG, 1024, (size_t)NSORT * 8, stream>>>(cntg, comp_s, comp_i,
                                                       topv, topi);
    k_nms<<<N_IMG, 1024, 0, stream>>>(topv, topi, locations, box_reg, imsz, out);
}